// HolographicBundle_82497731821662
// MI455X (gfx1250) — compile-verified
//
#include <hip/hip_runtime.h>

#define D      4096
#define DMASK  4095
#define BS     128
#define NPAIRS 64          // B(4) * nb(16)
#define HB_EPS 1e-8f

typedef __attribute__((ext_vector_type(2))) float v2f;
typedef __attribute__((ext_vector_type(8))) float v8f;

// ---------------- Kernel 1: softmax over the 128 position weights ----------
__global__ void hb_softmax128(const float* __restrict__ pw, float* __restrict__ w) {
    __shared__ float sm[BS];
    int t = threadIdx.x;                 // blockDim.x == 128
    float x = pw[t];
    sm[t] = x; __syncthreads();
    for (int s = BS / 2; s > 0; s >>= 1) {
        if (t < s) sm[t] = fmaxf(sm[t], sm[t + s]);
        __syncthreads();
    }
    float mx = sm[0]; __syncthreads();
    float e = expf(x - mx);
    sm[t] = e; __syncthreads();
    for (int s = BS / 2; s > 0; s >>= 1) {
        if (t < s) sm[t] += sm[t + s];
        __syncthreads();
    }
    w[t] = e / sm[0];
}

// ---- Kernel 2: y[pair,q] = sum_s w[s] * inputs[pair, s, (q+s) mod D] -------
// Bandwidth-dominant step: reads the full 134 MB input once, coalesced.
__global__ void __launch_bounds__(256) hb_shiftsum(
        const float* __restrict__ inp, const float* __restrict__ w,
        float* __restrict__ y) {
    __shared__ float sw[BS];
    int t = threadIdx.x;
    if (t < BS) sw[t] = w[t];
    __syncthreads();

    int pair = blockIdx.y;                       // 0..63
    int q0 = blockIdx.x * 512 + t;               // 8 chunks of 512 along D
    int q1 = q0 + 256;
    const float* prow = inp + (size_t)pair * BS * D;
    float acc0 = 0.0f, acc1 = 0.0f;
    #pragma unroll 4
    for (int s = 0; s < BS; ++s) {
        const float* row = prow + (size_t)s * D;
        float ws = sw[s];
        acc0 = fmaf(ws, row[(q0 + s) & DMASK], acc0);
        acc1 = fmaf(ws, row[(q1 + s) & DMASK], acc1);
    }
    y[pair * D + q0] = acc0;
    y[pair * D + q1] = acc1;
}

// ---- Kernel 3: bundle = y[64,4096] x circulant(base), fp32 WMMA ------------
// C[q,n] = base[(q-n) mod D]. One wave per 16x16 output tile, K = 4096.
// lbase holds base[] duplicated (2*D floats), so the per-lane column pointer
// bcol = lbase + ((2*half - n) mod D) can be indexed with a PLAIN offset k0:
// start + k0 + 1 <= 4095 + 4092 + 1 < 8192 -> no per-iteration modular mask,
// and the compiler folds k0 into immediate DS offsets.
__global__ void __launch_bounds__(256) hb_gemm_wmma(
        const float* __restrict__ y, const float* __restrict__ base,
        float* __restrict__ out) {
    __shared__ float lbase[2 * D];               // 32 KB
    int t = threadIdx.x;
    for (int i = t; i < D; i += 256) {
        float v = base[i];
        lbase[i]     = v;
        lbase[i + D] = v;
    }
    __syncthreads();

    int wave = t >> 5;
    int lane = t & 31;
    int half = lane >> 4;        // K-pair select for A/B fragments
    int l    = lane & 15;        // M row (A) / N col (B)

    int m0 = blockIdx.y * 16;                    // 4 M-tiles (64 rows)
    int n0 = blockIdx.x * 128 + wave * 16;       // 8 waves -> 128 cols/block
    int n  = n0 + l;
    const float* arow = y + (size_t)(m0 + l) * D;
    // B column base: hoisted modular start; k0 indexes linearly from here.
    const float* bcol = lbase + ((2 * half - n + 8192) & DMASK);

    v8f acc = {};
    for (int kc = 0; kc < D; kc += 64) {
        __builtin_prefetch(arow + kc + 64, 0, 0);   // global_prefetch_b8 on A stream
        #pragma unroll
        for (int kk = 0; kk < 64; kk += 4) {
            int k0 = kc + kk;
            // A 16x4 f32 fragment: lane l row m0+l, K = k0 + 2*half .. +1
            float2 af = *(const float2*)(arow + k0 + 2 * half);
            v2f a = { af.x, af.y };
            // B 4x16 fragment: B[K,n] = base[(K - n) mod D]
            v2f b = { bcol[k0], bcol[k0 + 1] };
            acc = __builtin_amdgcn_wmma_f32_16x16x4_f32(
                      /*neg_a=*/false, a, /*neg_b=*/false, b,
                      /*c_mod=*/(short)0, acc, /*reuse_a=*/false, /*reuse_b=*/false);
        }
    }
    // D layout: VGPR v -> row m0 + v + 8*half, col n
    #pragma unroll
    for (int v = 0; v < 8; ++v)
        out[(size_t)(m0 + v + 8 * half) * D + n] = acc[v];
}

// ---- Kernel 4: in-place row-wise L2 normalize (64 rows of 4096) ------------
__global__ void __launch_bounds__(256) hb_normalize(float* __restrict__ out) {
    __shared__ float sm[256];
    int t = threadIdx.x;
    float* rowp = out + (size_t)blockIdx.x * D;
    float v[16];
    float ssq = 0.0f;
    #pragma unroll
    for (int i = 0; i < 16; ++i) {
        v[i] = rowp[t + i * 256];
        ssq = fmaf(v[i], v[i], ssq);
    }
    sm[t] = ssq; __syncthreads();
    for (int s = 128; s > 0; s >>= 1) {
        if (t < s) sm[t] += sm[t + s];
        __syncthreads();
    }
    float scale = rsqrtf(fmaxf(sm[0], HB_EPS));
    #pragma unroll
    for (int i = 0; i < 16; ++i)
        rowp[t + i * 256] = v[i] * scale;
}

extern "C" void kernel_launch(void* const* d_in, const int* in_sizes, int n_in,
                              void* d_out, int out_size, void* d_ws, size_t ws_size,
                              hipStream_t stream) {
    (void)in_sizes; (void)n_in; (void)out_size; (void)ws_size;
    const float* inp  = (const float*)d_in[0];   // [4, 2048, 4096] f32
    const float* base = (const float*)d_in[1];   // [4096] f32
    const float* pw   = (const float*)d_in[2];   // [128] f32
    float* out = (float*)d_out;                  // [4, 16, 4096] f32

    float* w = (float*)d_ws;                           // 128 floats
    float* y = (float*)((char*)d_ws + 1024);           // 64*4096 floats (1 MB)

    hb_softmax128<<<1, 128, 0, stream>>>(pw, w);
    hb_shiftsum  <<<dim3(8, NPAIRS), 256, 0, stream>>>(inp, w, y);
    hb_gemm_wmma <<<dim3(32, 4),     256, 0, stream>>>(y, base, out);
    hb_normalize <<<NPAIRS,          256, 0, stream>>>(out);
}